// FeatureRender_75866302316616
// MI455X (gfx1250) — compile-verified
//
#include <hip/hip_runtime.h>

// FeatureRender: DensePose-style texture mapping (pure gather/select, memory bound).
//
// Shapes (from setup_inputs):
//   source_feature / target_feature : [4,16,800,1200] f32
//   dense_pose                      : [4,512,512,3]   f32  (cls, U, V)
//   source_texture                  : [4,3,800,1200]  f32
//   target_image                    : [4,3,512,512]   f32
//   output                          : [4,19,512,512]  f32  (16 feat ch + 3 apparel ch)

#define B_      4
#define C_      16
#define H_      800
#define W_      1200
#define HF_     200      // H/4
#define WF_     200      // W/6
#define PLANE_  (H_*W_)  // 960000 elems per (b,c) feature plane
#define DHW_    (512*512)
#define NPIX_   (B_*DHW_)
#define OUTC_   19

// Parts taken from source_feature in the union: {1, 14..21}  -> bitmask
#define SRC_PART_MASK 0x003FC002u

__global__ __launch_bounds__(256) void feature_render_kernel(
    const float* __restrict__ src_feat,   // [4,16,800,1200]
    const float* __restrict__ tgt_feat,   // [4,16,800,1200]
    const float* __restrict__ pose,       // [4,512,512,3]
    const float* __restrict__ src_tex,    // [4,3,800,1200]
    const float* __restrict__ tgt_img,    // [4,3,512,512]
    float*       __restrict__ out)        // [4,19,512,512]
{
    int idx = blockIdx.x * blockDim.x + threadIdx.x;
    if (idx >= NPIX_) return;

    const int b   = idx >> 18;        // / 262144
    const int pix = idx & (DHW_ - 1); // plane-local offset (== y*512 + x)

    // ---- decode dense_pose (contiguous 12B per pixel; compiler can fuse to b96)
    const float* pp = pose + (size_t)idx * 3;
    const float clsf = pp[0];
    const float U    = pp[1];
    const float V    = pp[2];

    const int  cls   = (int)clsf;                       // trunc == astype(int32), cls >= 0
    const bool bg    = (cls != 0);
    const bool valid = (cls >= 1) && (V != 0.0f);

    int p = cls - 1;
    p = p < 0 ? 0 : (p > 23 ? 23 : p);

    // trunc-toward-zero matches .astype(int32) for non-negative values
    const int ui = (int)((U * 199.0f) / 255.0f);                 // U * (hf-1) / 255
    const int vi = (int)(((255.0f - V) * 199.0f) / 255.0f);      // (255-V)*(wf-1)/255

    // unfold inverse: part p = tile(p/6, p%6) of the [800,1200] plane
    const int tr  = p / 6;
    const int tc  = p - tr * 6;
    const int off = (tr * HF_ + ui) * W_ + (tc * WF_ + vi);      // gather offset within a plane

    const size_t outBase = (size_t)b * OUTC_ * DHW_ + pix;

    // ---- 16 mapped_feature channels: gather from union atlas (base-ptr select per part)
    if (valid) {
        const bool fromSrc = (SRC_PART_MASK >> p) & 1u;
        const float* fp = (fromSrc ? src_feat : tgt_feat)
                        + (size_t)b * C_ * PLANE_ + off;
        float v[C_];
#pragma unroll
        for (int c = 0; c < C_; ++c)                 // 16 independent global_load_b32 (one wait)
            v[c] = fp[(size_t)c * PLANE_];
#pragma unroll
        for (int c = 0; c < C_; ++c)                 // write-once output -> TH=NT stores
            __builtin_nontemporal_store(v[c], out + outBase + (size_t)c * DHW_);
    } else {
#pragma unroll
        for (int c = 0; c < C_; ++c)
            __builtin_nontemporal_store(0.0f, out + outBase + (size_t)c * DHW_);
    }

    // ---- 3 mapped_apparel channels
    const bool apparel = (cls == 2) || (cls >= 15 && cls <= 22);
    float a3[3] = {0.0f, 0.0f, 0.0f};
    if (bg) {
        if (apparel) {
            if (valid) {                              // mapped_src = gather * valid
                const float* tp = src_tex + (size_t)b * 3 * PLANE_ + off;
#pragma unroll
                for (int c = 0; c < 3; ++c)
                    a3[c] = tp[(size_t)c * PLANE_];
            }
            // else: gather would be multiplied by valid==0 -> stays 0
        } else {                                      // identity region: coalesced, read-once
            const float* ip = tgt_img + (size_t)b * 3 * DHW_ + pix;
#pragma unroll
            for (int c = 0; c < 3; ++c)
                a3[c] = __builtin_nontemporal_load(ip + (size_t)c * DHW_);
        }
    }
#pragma unroll
    for (int c = 0; c < 3; ++c)
        __builtin_nontemporal_store(a3[c], out + outBase + (size_t)(C_ + c) * DHW_);
}

extern "C" void kernel_launch(void* const* d_in, const int* in_sizes, int n_in,
                              void* d_out, int out_size, void* d_ws, size_t ws_size,
                              hipStream_t stream) {
    (void)in_sizes; (void)n_in; (void)out_size; (void)d_ws; (void)ws_size;
    const float* src_feat = (const float*)d_in[0];  // source_feature
    const float* tgt_feat = (const float*)d_in[1];  // target_feature
    const float* pose     = (const float*)d_in[2];  // dense_pose
    const float* src_tex  = (const float*)d_in[3];  // source_texture
    const float* tgt_img  = (const float*)d_in[4];  // target_image
    float* out = (float*)d_out;

    const int threads = 256;                 // 8 wave32 waves per block
    const int blocks  = NPIX_ / threads;     // 4096 exactly
    feature_render_kernel<<<blocks, threads, 0, stream>>>(
        src_feat, tgt_feat, pose, src_tex, tgt_img, out);
}